// SegmentEmbedder_36593121362278
// MI455X (gfx1250) — compile-verified
//
#include <hip/hip_runtime.h>

typedef float v2f __attribute__((ext_vector_type(2)));
typedef float v8f __attribute__((ext_vector_type(8)));

constexpr int      kDim    = 64;
constexpr int      kSegs   = 65536;
constexpr unsigned kTokens = 2097152u;
constexpr int      kChunk  = 256;                       // tokens per wave
constexpr float    kOffW   = 0.69314718055994530942f;   // log1p(1.0)

// ---------------------------------------------------------------------------
// Kernel 1: zero the output accumulators and the per-segment weight sums.
// ---------------------------------------------------------------------------
__global__ __launch_bounds__(256) void zero_ws_out(float4* __restrict__ out4,
                                                   float4* __restrict__ ws4,
                                                   int nOut4, int nWs4) {
  int i = blockIdx.x * blockDim.x + threadIdx.x;
  float4 z = make_float4(0.f, 0.f, 0.f, 0.f);
  if (i < nOut4) {
    out4[i] = z;
  } else {
    int j = i - nOut4;
    if (j < nWs4) ws4[j] = z;
  }
}

// ---------------------------------------------------------------------------
// Kernel 2: WMMA-based weighted segmented sum.
// Each wave owns kChunk contiguous (sorted-by-segment) tokens.
// 16-token subgroups -> at most 16 distinct segments -> 16 slots (M rows).
// A[16x4] carries selection*weight, B[4x16] carries embedding dims,
// C[16x16] f32 accumulates exact fp32 weighted sums; 4 tiles cover D=64.
// ---------------------------------------------------------------------------
__global__ __launch_bounds__(256) void seg_accum_wmma(
    const float* __restrict__ E, const float* __restrict__ inten,
    const int* __restrict__ segid, float* __restrict__ out,
    float* __restrict__ wsum)
{
  const int lane = (int)(threadIdx.x & 31u);
  const int wave = (int)((blockIdx.x * blockDim.x + threadIdx.x) >> 5);
  const unsigned tok0 = (unsigned)wave * (unsigned)kChunk;
  const bool hi  = lane >= 16;
  const int  n   = lane & 15;      // tile column (dim) / A-matrix row M
  const int  kmy = hi ? 2 : 0;     // K pair this lane holds in A/B layout

  const v8f vzero = {0.f, 0.f, 0.f, 0.f, 0.f, 0.f, 0.f, 0.f};
  v8f c0 = vzero, c1 = vzero, c2 = vzero, c3 = vzero;

  for (unsigned base = tok0; base < tok0 + (unsigned)kChunk; base += 32u) {
    const int   seg = segid[base + (unsigned)lane];
    const float w   = log1pf(inten[base + (unsigned)lane]);

    // New-segment flag within each 16-token subgroup (ids are sorted).
    const int  segPrev = __shfl_up(seg, 1);
    const bool flag    = (n == 0) || (seg != segPrev);
    const unsigned bal     = (unsigned)__ballot(flag);
    const unsigned subMask = hi ? 0xFFFF0000u : 0x0000FFFFu;
    const unsigned leMask  = (2u << lane) - 1u;
    const int slot    = __popc(bal & subMask & leMask) - 1;   // 0..15
    const int countLo = __popc(bal & 0x0000FFFFu);
    const int countHi = __popc(bal >> 16);

    // slot -> global segment id map (low subgroup lanes 0-15, high 16-31).
    // All writers to a given dst lane carry the SAME seg value, so the
    // scatter collision is benign; unused lanes are masked by `count`.
    const int map = __builtin_amdgcn_ds_permute((slot + (hi ? 16 : 0)) << 2, seg);

    #pragma unroll
    for (int s = 0; s < 2; ++s) {
      const int count = s ? countHi : countLo;
      float wpart = 0.f;

      #pragma unroll
      for (int g = 0; g < 4; ++g) {
        const int tl = s * 16 + g * 4 + kmy;          // token lane for K=kmy
        const float w0 = __shfl(w, tl);
        const float w1 = __shfl(w, tl + 1);
        const int   s0 = __shfl(slot, tl);
        const int   s1 = __shfl(slot, tl + 1);
        v2f a;
        a.x = (s0 == n) ? w0 : 0.f;                   // A[M=n][K=kmy]
        a.y = (s1 == n) ? w1 : 0.f;                   // A[M=n][K=kmy+1]
        wpart += a.x + a.y;                           // per-slot weight sum

        const float* r0 = E + (size_t)(base + (unsigned)tl) * kDim + n;
        const float* r1 = r0 + kDim;
        v2f b0, b1, b2, b3;
        b0.x = __builtin_nontemporal_load(r0 +  0);   // B[K=kmy][n], tile 0
        b0.y = __builtin_nontemporal_load(r1 +  0);   // B[K=kmy+1][n]
        b1.x = __builtin_nontemporal_load(r0 + 16);
        b1.y = __builtin_nontemporal_load(r1 + 16);
        b2.x = __builtin_nontemporal_load(r0 + 32);
        b2.y = __builtin_nontemporal_load(r1 + 32);
        b3.x = __builtin_nontemporal_load(r0 + 48);
        b3.y = __builtin_nontemporal_load(r1 + 48);

        c0 = __builtin_amdgcn_wmma_f32_16x16x4_f32(false, a, false, b0, (short)0, c0, false, false);
        c1 = __builtin_amdgcn_wmma_f32_16x16x4_f32(false, a, false, b1, (short)0, c1, false, false);
        c2 = __builtin_amdgcn_wmma_f32_16x16x4_f32(false, a, false, b2, (short)0, c2, false, false);
        c3 = __builtin_amdgcn_wmma_f32_16x16x4_f32(false, a, false, b3, (short)0, c3, false, false);
      }

      // ---- flush this subgroup's slots ----
      const float wfull = wpart + __shfl_xor(wpart, 16);
      const int   mapN  = __shfl(map, n + s * 16);
      if (!hi && n < count) atomicAdd(&wsum[mapN], wfull);

      #pragma unroll
      for (int r = 0; r < 8; ++r) {
        const int m    = r + (hi ? 8 : 0);            // C/D row M for VGPR r
        const int segm = __shfl(map, m + s * 16);
        if (m < count) {
          float* o = out + (size_t)segm * kDim + n;
          atomicAdd(o +  0, c0[r]);
          atomicAdd(o + 16, c1[r]);
          atomicAdd(o + 32, c2[r]);
          atomicAdd(o + 48, c3[r]);
        }
      }
      c0 = vzero; c1 = vzero; c2 = vzero; c3 = vzero;
    }
  }
}

// ---------------------------------------------------------------------------
// Kernel 3: add offset-token contribution and L1-normalize.
// ---------------------------------------------------------------------------
__global__ __launch_bounds__(256) void seg_finalize(float4* __restrict__ out4,
                                                    const float* __restrict__ wsum,
                                                    const float4* __restrict__ off4)
{
  const int i = blockIdx.x * blockDim.x + threadIdx.x;  // over kSegs * 16
  const int b = i >> 4;
  const int d = i & 15;
  float4 o = out4[i];
  const float4 f = off4[d];
  const float inv = 1.f / (wsum[b] + kOffW);
  o.x = (o.x + kOffW * f.x) * inv;
  o.y = (o.y + kOffW * f.y) * inv;
  o.z = (o.z + kOffW * f.z) * inv;
  o.w = (o.w + kOffW * f.w) * inv;
  out4[i] = o;
}

extern "C" void kernel_launch(void* const* d_in, const int* in_sizes, int n_in,
                              void* d_out, int out_size, void* d_ws, size_t ws_size,
                              hipStream_t stream) {
  (void)in_sizes; (void)n_in; (void)out_size; (void)ws_size;
  const float* E     = (const float*)d_in[0];   // [T, 64]
  const float* inten = (const float*)d_in[1];   // [T]
  const int*   segid = (const int*)d_in[2];     // [T], sorted
  const float* offt  = (const float*)d_in[3];   // [1, 64]
  float* out  = (float*)d_out;                  // [B, 64]
  float* wsum = (float*)d_ws;                   // [B] scratch

  const int nOut4 = kSegs * kDim / 4;           // 1,048,576
  const int nWs4  = kSegs / 4;                  // 16,384
  const int zT    = nOut4 + nWs4;
  zero_ws_out<<<(zT + 255) / 256, 256, 0, stream>>>((float4*)out, (float4*)wsum,
                                                    nOut4, nWs4);

  const int waves  = (int)(kTokens / (unsigned)kChunk);  // 8192
  const int blocks = waves / 8;                          // 1024 blocks x 8 waves
  seg_accum_wmma<<<blocks, 256, 0, stream>>>(E, inten, segid, out, wsum);

  seg_finalize<<<(kSegs * 16) / 256, 256, 0, stream>>>((float4*)out, wsum,
                                                       (const float4*)offt);
}